// RMSNormSparse_45011257262542
// MI455X (gfx1250) — compile-verified
//
#include <hip/hip_runtime.h>
#include <stdint.h>

// Masked RMSNorm: out = weight * (mask?x:0) * rsqrt(mean((mask?x:0)^2) + eps)
// Shapes fixed by the reference: rows = 4*4096 = 16384, D = 4096 (f32).
// Memory-bound (~0.55 flop/byte) -> optimize the data-movement path:
//   - b128 NT loads/stores for x/out (stream past the 192MB L2)
//   - CDNA5 async-to-LDS engine for the 1-byte mask row (ASYNCcnt)
//   - wave32 shuffle reduction (gfx1250 is wave32-only)

#define D      4096
#define BLOCK  256
#define CHUNKS 4            // BLOCK * CHUNKS * 4 floats == D
#define EPS    1e-5f

typedef float v4f __attribute__((ext_vector_type(4)));

__device__ __forceinline__ uint32_t lds_lo32(const void* p) {
  // generic pointer to __shared__: low 32 bits are the LDS byte offset
  return (uint32_t)(uintptr_t)p;
}

__global__ __launch_bounds__(BLOCK) void rmsnorm_sparse_kernel(
    const float* __restrict__ x, const unsigned char* __restrict__ mask,
    const float* __restrict__ w, float* __restrict__ out, int nrows) {
  __shared__ unsigned int smem_mask[D / 4];   // 4 KB: one mask row, 4 bytes/elem-group
  __shared__ float s_partial[BLOCK / 32];     // one partial per wave (8 waves)
  __shared__ float s_rinv;

  const int row = blockIdx.x;
  if (row >= nrows) return;
  const int t = threadIdx.x;

  const float*         xrow = x    + (size_t)row * D;
  float*               orow = out  + (size_t)row * D;
  const unsigned char* mrow = mask + (size_t)row * D;

  // ---- stage mask row into LDS via the CDNA5 async engine (ASYNCcnt) ----
  // GVS addressing: global = SGPR64(base) + VGPR32(offset); LDS = VGPR32(addr)
  {
    unsigned long long sbase = (unsigned long long)(uintptr_t)mrow;
#pragma unroll
    for (int c = 0; c < CHUNKS; ++c) {
      unsigned int voff  = (unsigned int)((c * BLOCK + t) * 4);
      unsigned int laddr = lds_lo32(&smem_mask[c * BLOCK + t]);
      asm volatile("global_load_async_to_lds_b32 %0, %1, %2"
                   :
                   : "v"(laddr), "v"(voff), "s"(sbase)
                   : "memory");
    }
  }

  // ---- load x into registers (b128, non-temporal) while async engine runs ----
  v4f xv[CHUNKS];
#pragma unroll
  for (int c = 0; c < CHUNKS; ++c) {
    xv[c] = __builtin_nontemporal_load((const v4f*)(xrow + (size_t)(c * BLOCK + t) * 4));
  }

  // wait for this wave's async mask transfers; lanes read back only their own words
  asm volatile("s_wait_asynccnt 0x0" ::: "memory");

  // ---- apply mask, accumulate sum of squares ----
  float ss = 0.0f;
#pragma unroll
  for (int c = 0; c < CHUNKS; ++c) {
    unsigned int m = smem_mask[c * BLOCK + t];
    v4f v = xv[c];
    float a0 = (m & 0x000000ffu) ? v.x : 0.0f;
    float a1 = (m & 0x0000ff00u) ? v.y : 0.0f;
    float a2 = (m & 0x00ff0000u) ? v.z : 0.0f;
    float a3 = (m & 0xff000000u) ? v.w : 0.0f;
    v.x = a0; v.y = a1; v.z = a2; v.w = a3;
    xv[c] = v;
    ss = fmaf(a0, a0, ss);
    ss = fmaf(a1, a1, ss);
    ss = fmaf(a2, a2, ss);
    ss = fmaf(a3, a3, ss);
  }

  // ---- wave32 tree reduction ----
#pragma unroll
  for (int off = 16; off > 0; off >>= 1)
    ss += __shfl_xor(ss, off, 32);

  const int wave = t >> 5;
  const int lane = t & 31;
  if (lane == 0) s_partial[wave] = ss;
  __syncthreads();

  if (t == 0) {
    float tot = 0.0f;
#pragma unroll
    for (int i = 0; i < BLOCK / 32; ++i) tot += s_partial[i];
    float variance = tot * (1.0f / (float)D);
    s_rinv = rsqrtf(variance + EPS);
  }
  __syncthreads();
  const float rinv = s_rinv;

  // ---- scale by weight and rinv, b128 non-temporal stores ----
#pragma unroll
  for (int c = 0; c < CHUNKS; ++c) {
    const size_t base = (size_t)(c * BLOCK + t) * 4;
    v4f wv = *(const v4f*)(w + base);   // weight: L2-resident, regular load
    v4f v  = xv[c];
    v4f o;
    o.x = wv.x * v.x * rinv;
    o.y = wv.y * v.y * rinv;
    o.z = wv.z * v.z * rinv;
    o.w = wv.w * v.w * rinv;
    __builtin_nontemporal_store(o, (v4f*)(orow + base));
  }
}

extern "C" void kernel_launch(void* const* d_in, const int* in_sizes, int n_in,
                              void* d_out, int out_size, void* d_ws, size_t ws_size,
                              hipStream_t stream) {
  const float*         x    = (const float*)d_in[0];
  const unsigned char* mask = (const unsigned char*)d_in[1];  // jnp bool = 1 byte
  const float*         w    = (const float*)d_in[2];
  float*               out  = (float*)d_out;

  const int nrows = in_sizes[0] / D;  // 16384 for the reference shapes
  rmsnorm_sparse_kernel<<<nrows, BLOCK, 0, stream>>>(x, mask, w, out, nrows);
}